// kernel_generated_3_21887153341263
// MI455X (gfx1250) — compile-verified
//
#include <hip/hip_runtime.h>

// D = A×B + C via V_WMMA_F32_16X16X4_F32 (native fp32 WMMA on gfx1250).
typedef float v2f __attribute__((ext_vector_type(2)));
typedef float v8f __attribute__((ext_vector_type(8)));

namespace {
constexpr int CIN        = 512;   // in-channels per group
constexpr int COUT       = 512;   // out-channels per group
constexpr int HW         = 49;    // 7x7 spatial
constexpr int LDS_STRIDE = 50;    // 49 data cols + 1 zero col per row
}

__global__ __launch_bounds__(128)
void grouped_conv_roll_wmma(const float* __restrict__ x,
                            const float* __restrict__ w,
                            float* __restrict__ out) {
  extern __shared__ float ldsx[];   // [CIN][LDS_STRIDE], 102400 B

  const int tid  = threadIdx.x;
  const int lane = tid & 31;
  const int wave = tid >> 5;                   // 0..3
  const int o    = blockIdx.x >> 3;            // group 0/1
  const int k0   = ((blockIdx.x & 7) * 4 + wave) * 16;  // this wave's out-ch tile

  // ---- Stage this group's activation slab into LDS (row stride 50, zero col 49)
  const float* xg = x + o * CIN * HW;
  for (int r = tid; r < CIN; r += 128) {
    const float* src = xg + r * HW;
    float* dst = ldsx + r * LDS_STRIDE;
    #pragma unroll 7
    for (int c = 0; c < HW; ++c) dst[c] = src[c];
    dst[HW] = 0.0f;                            // zero pad slot for this row
  }
  __syncthreads();

  // ---- Per-lane fragment coordinates
  const int ncol = lane & 15;                  // N column within a 16-wide tile
  const int half = lane >> 4;                  // 0: lanes 0-15, 1: lanes 16-31
  const int koff = half * 2;                   // K sub-offset for A/B fragments

  // Precompute B spatial offsets per (tap j, N-tile t). Invalid / padded
  // columns point at the zero column (index 49) of the row -> loads read 0.
  int boff[3][4];
  #pragma unroll
  for (int j = 0; j < 3; ++j) {
    #pragma unroll
    for (int t = 0; t < 4; ++t) {
      const int nm = t * 16 + ncol;
      int off = HW;                            // zero column
      if (nm < HW) {
        const int n = nm / 7, m = nm % 7;
        const int p = m + j;                   // padded width index
        if (p >= 1 && p <= 7) off = n * 7 + (p + 5) % 7;  // input roll folded in
      }
      boff[j][t] = off;
    }
  }

  v8f acc[4] = {};                             // 16x64 fp32 output tile

  // w[i][k][j] flat: i*1536 + k*3 + j ; this lane reads column k = k0+ncol
  const float* wk = w + (k0 + ncol) * 3;

  for (int j = 0; j < 3; ++j) {
    const int o0 = boff[j][0], o1 = boff[j][1];
    const int o2 = boff[j][2], o3 = boff[j][3];
    for (int i0 = 0; i0 < CIN; i0 += 4) {      // K step of 4 (fp32 WMMA depth)
      const int ia = i0 + koff;
      // A fragment: A[M=k][K=i] = w[i, k0+M, j]
      v2f A = { wk[ia * (COUT * 3) + j],
                wk[(ia + 1) * (COUT * 3) + j] };
      // B fragments from LDS: B[K=i][N] = xpad[o, i, spatial(N, j)]
      const float* br0 = ldsx + ia * LDS_STRIDE;
      const float* br1 = br0 + LDS_STRIDE;
      v2f B0 = { br0[o0], br1[o0] };
      v2f B1 = { br0[o1], br1[o1] };
      v2f B2 = { br0[o2], br1[o2] };
      v2f B3 = { br0[o3], br1[o3] };
      acc[0] = __builtin_amdgcn_wmma_f32_16x16x4_f32(false, A, false, B0, (short)0, acc[0], false, false);
      acc[1] = __builtin_amdgcn_wmma_f32_16x16x4_f32(false, A, false, B1, (short)0, acc[1], false, false);
      acc[2] = __builtin_amdgcn_wmma_f32_16x16x4_f32(false, A, false, B2, (short)0, acc[2], false, false);
      acc[3] = __builtin_amdgcn_wmma_f32_16x16x4_f32(false, A, false, B3, (short)0, acc[3], false, false);
    }
  }

  // ---- Store with the final roll along H (out row = (n+1)%7).
  // C/D layout: VGPR v -> M = v (lanes 0-15) or v+8 (lanes 16-31), N = lane%16.
  float* outg = out + o * COUT * HW;
  const int kbase = k0 + half * 8;
  #pragma unroll
  for (int t = 0; t < 4; ++t) {
    const int nm = t * 16 + ncol;
    if (nm < HW) {
      const int n = nm / 7, m = nm % 7;
      const int sp = ((n + 1) % 7) * 7 + m;
      #pragma unroll
      for (int v = 0; v < 8; ++v)
        outg[(kbase + v) * HW + sp] = acc[t][v];
    }
  }
}

extern "C" void kernel_launch(void* const* d_in, const int* in_sizes, int n_in,
                              void* d_out, int out_size, void* d_ws, size_t ws_size,
                              hipStream_t stream) {
  (void)in_sizes; (void)n_in; (void)d_ws; (void)ws_size; (void)out_size;
  const float* x = (const float*)d_in[0];   // (1,1024,7,7) f32
  const float* w = (const float*)d_in[1];   // (512,512,3)  f32
  // d_in[2] = shift (==1) is folded into the kernel's index arithmetic.
  float* out = (float*)d_out;               // (1,1024,7,7) f32

  dim3 grid(16), block(128);
  size_t shmem = (size_t)CIN * LDS_STRIDE * sizeof(float);  // 102400 B (<320KB/WGP)
  hipLaunchKernelGGL(grouped_conv_roll_wmma, grid, block, shmem, stream, x, w, out);
}